// MinibatchDiscrimination_12850542149933
// MI455X (gfx1250) — compile-verified
//
#include <hip/hip_runtime.h>

// MinibatchDiscrimination for MI455X (gfx1250, wave32).
//
// Phase 1: C[256,2048] = x[256,512] @ W[512,2048] via V_WMMA_F32_16X16X4_F32
//          (fp32 matrix path: faithful precision; GEMM is only 0.54 GFLOP so
//          the f32 WMMA rate is more than enough — problem is L2-latency bound).
// Phase 2: o[b,f] = sum_a exp(-sum_k |C[a,f*16+k] - C[b,f*16+k]|), fused with
//          the copy of x into out[:, :512]. All operands are L2-resident
//          (total footprint ~7 MB << 192 MB L2).

typedef float v2f __attribute__((ext_vector_type(2)));
typedef float v8f __attribute__((ext_vector_type(8)));

#define IN_F     512
#define OUT_F    128
#define KD       16
#define BATCH    256
#define NCOL     (OUT_F * KD)      // 2048
#define OUT_COLS (IN_F + OUT_F)    // 640

// ---------------------------------------------------------------------------
// GEMM: one wave32 per 16x16 output tile; K loop in steps of 4 (f32 WMMA).
// A 16x4 layout: lane<16 -> (K=kk,kk+1) of row M=lane ; lane>=16 -> (K=kk+2,kk+3)
//   => contiguous float2 load from row-major x.
// B 4x16 layout: VGPR0 = row (kk+2*half), VGPR1 = row (kk+2*half+1), N = lane&15
//   => coalesced loads from W (= T viewed as [512, 2048]).
// ---------------------------------------------------------------------------
__global__ __launch_bounds__(32) void mbd_gemm_f32_wmma(
    const float* __restrict__ x,   // [BATCH, IN_F]
    const float* __restrict__ W,   // [IN_F, NCOL]  (T reshaped, contiguous)
    float* __restrict__ C)         // [BATCH, NCOL]
{
  const int tile = blockIdx.x;          // 16 M-tiles * 128 N-tiles = 2048
  const int tm   = (tile & 15) << 4;    // M tile origin
  const int tn   = (tile >> 4) << 4;    // N tile origin
  const int lane = threadIdx.x;         // 0..31
  const int half = lane >> 4;           // 0: K pair (0,1), 1: K pair (2,3)
  const int l    = lane & 15;

  const float* arow = x + (tm + l) * IN_F + 2 * half;          // float2 per step
  const float* bcol = W + (2 * half) * NCOL + tn + l;          // 2 rows per step

  v8f acc = {};
#pragma unroll 4
  for (int kk = 0; kk < IN_F; kk += 4) {
    v2f a = *(const v2f*)(arow + kk);
    v2f b;
    b.x = bcol[kk * NCOL];
    b.y = bcol[kk * NCOL + NCOL];
    // (neg_a, A, neg_b, B, c_mod, C, reuse_a, reuse_b)
    acc = __builtin_amdgcn_wmma_f32_16x16x4_f32(
        false, a, false, b, (short)0, acc, false, false);
  }

  // D layout: VGPR r, lane l, half h -> C[tm + r + 8*h][tn + l]
  float* crow = C + (tm + 8 * half) * NCOL + tn + l;
#pragma unroll
  for (int r = 0; r < 8; ++r) crow[r * NCOL] = acc[r];
}

// ---------------------------------------------------------------------------
// Pairwise exp(-L1) reduction + fused copy of x into the output.
// One block per batch row b; thread f (0..127) owns feature f.
// ---------------------------------------------------------------------------
__global__ __launch_bounds__(128) void mbd_pair_exp(
    const float* __restrict__ M,   // [BATCH, NCOL]
    const float* __restrict__ x,   // [BATCH, IN_F]
    float* __restrict__ out)       // [BATCH, OUT_COLS]
{
  const int b = blockIdx.x;
  const int f = threadIdx.x;

  // Fused concat: copy x[b, :] -> out[b, :512] (128 float4 = 512 floats).
  const float4* xr = (const float4*)(x + b * IN_F);
  float4* orow = (float4*)(out + b * OUT_COLS);
  orow[f] = xr[f];

  // Keep m[b, f, 0:16] in registers (4 x float4 = 64 B).
  const float4* mb4 = (const float4*)(M + b * NCOL + f * KD);
  const float4 b0 = mb4[0], b1 = mb4[1], b2 = mb4[2], b3 = mb4[3];

  float acc = 0.0f;
  for (int a = 0; a < BATCH; ++a) {
    const float4* ma4 = (const float4*)(M + a * NCOL + f * KD);
    const float4 a0 = ma4[0], a1 = ma4[1], a2 = ma4[2], a3 = ma4[3];
    float s;
    s  = fabsf(a0.x - b0.x) + fabsf(a0.y - b0.y) + fabsf(a0.z - b0.z) + fabsf(a0.w - b0.w);
    s += fabsf(a1.x - b1.x) + fabsf(a1.y - b1.y) + fabsf(a1.z - b1.z) + fabsf(a1.w - b1.w);
    s += fabsf(a2.x - b2.x) + fabsf(a2.y - b2.y) + fabsf(a2.z - b2.z) + fabsf(a2.w - b2.w);
    s += fabsf(a3.x - b3.x) + fabsf(a3.y - b3.y) + fabsf(a3.z - b3.z) + fabsf(a3.w - b3.w);
    acc += __expf(-s);   // v_exp_f32 (trans unit), co-executes with VALU
  }
  out[b * OUT_COLS + IN_F + f] = acc;
}

extern "C" void kernel_launch(void* const* d_in, const int* in_sizes, int n_in,
                              void* d_out, int out_size, void* d_ws, size_t ws_size,
                              hipStream_t stream) {
  const float* x = (const float*)d_in[0];        // [256, 512]
  const float* T = (const float*)d_in[1];        // [512, 128, 16] == [512, 2048]
  float* out = (float*)d_out;                    // [256, 640]
  float* Cmat = (float*)d_ws;                    // [256, 2048] fp32 = 2 MB scratch

  // 2048 tiles, one wave32 each.
  mbd_gemm_f32_wmma<<<dim3(16 * 128), dim3(32), 0, stream>>>(x, T, Cmat);
  // 256 blocks x 128 threads (4 wave32s each).
  mbd_pair_exp<<<dim3(BATCH), dim3(128), 0, stream>>>(Cmat, x, out);
}